// fbm_dropout_18580028522890
// MI455X (gfx1250) — compile-verified
//
#include <hip/hip_runtime.h>
#include <hip/hip_bf16.h>
#include <stdint.h>

#define INPUT_SIZE     4096
#define N_GRID         64
#define PTS_PER_EPOCH  100
#define N_FIBERS       64

#define TB             256
#define COLS_PER_TILE  1024   // TB threads * 4 floats
#define ROWS_PER_BLOCK 16

typedef float vfloat4 __attribute__((ext_vector_type(4)));
typedef int   vint4   __attribute__((ext_vector_type(4)));

// ---------------------------------------------------------------------------
// Kernel 1: mask[i] = 1.0 for all 4096 cells.
// ---------------------------------------------------------------------------
__global__ void mask_init_kernel(float* __restrict__ mask) {
    int i = blockIdx.x * blockDim.x + threadIdx.x;
    if (i < INPUT_SIZE) mask[i] = 1.0f;
}

// ---------------------------------------------------------------------------
// Kernel 2: for each of the 6400 epoch points, find its unique candidate cell
// (rectangles are disjoint) and write 0.0 if the point lies inside the
// central-half rectangle. Idempotent 0.0 writes -> no atomics needed.
// ---------------------------------------------------------------------------
__global__ void mask_scatter_kernel(const float* __restrict__ fx,
                                    const float* __restrict__ fy,
                                    const int* __restrict__ epoch,
                                    int total_pts,
                                    float* __restrict__ mask) {
    const int p = blockIdx.x * blockDim.x + threadIdx.x;  // 0..6399
    const int P = N_FIBERS * PTS_PER_EPOCH;
    if (p >= P) return;

    const int t0 = epoch[0] * PTS_PER_EPOCH;
    const int f  = p / PTS_PER_EPOCH;
    const int j  = p - f * PTS_PER_EPOCH;

    const float px = fx[f * total_pts + t0 + j];
    const float py = fy[f * total_pts + t0 + j];

    int gx = (int)floorf(px * (float)N_GRID);
    int gy = (int)floorf(py * (float)N_GRID);
    gx = min(max(gx, 0), N_GRID - 1);
    gy = min(max(gy, 0), N_GRID - 1);

    // size = 1/64 and gap = 1/256 are exact dyadic fp32 values, so these
    // bounds match the reference bit-for-bit.
    const float size = 1.0f / (float)N_GRID;
    const float gap  = size * 0.25f;
    const float xl = (float)gx * size + gap;
    const float xh = (float)gx * size + 3.0f * gap;
    const float yl = (float)gy * size + gap;
    const float yh = (float)gy * size + 3.0f * gap;

    if (px >= xl && px <= xh && py >= yl && py <= yh)
        mask[gy * N_GRID + gx] = 0.0f;
}

// ---------------------------------------------------------------------------
// Kernel 3: out[r][c] = in[r][c] * mask[c].  Pure streaming: 268 MB of
// HBM traffic -> ~11.5 us roofline at 23.3 TB/s.  Each block async-copies its
// 4 KB mask slice into LDS once (gfx1250 async global->LDS path, ASYNCcnt),
// pulls it into registers, then streams ROWS_PER_BLOCK rows with b128
// non-temporal loads/stores (read/write-once data: keep L2 clean).
// ---------------------------------------------------------------------------
__global__ __launch_bounds__(TB)
void mask_mul_kernel(const float* __restrict__ in,
                     const float* __restrict__ mask,
                     float* __restrict__ out) {
    __shared__ float lmask[COLS_PER_TILE];

    const int tid  = threadIdx.x;
    const int col0 = blockIdx.x * COLS_PER_TILE + tid * 4;   // this thread's 4 cols
    const int row0 = blockIdx.y * ROWS_PER_BLOCK;

    // Async copy: each lane moves 16B of mask global->LDS (one
    // global_load_async_to_lds_b128 per wave; 8 waves cover the 4 KB tile).
    __builtin_amdgcn_global_load_async_to_lds_b128(
        (vint4*)(mask + col0), (vint4*)&lmask[tid * 4], 0, 0);
    __builtin_amdgcn_s_wait_asynccnt(0);
    __syncthreads();

    const vfloat4 m = *(const vfloat4*)&lmask[tid * 4];

    const size_t cstride4 = INPUT_SIZE / 4;
    size_t idx = (size_t)row0 * cstride4 + (size_t)(col0 >> 2);
    const vfloat4* __restrict__ in4  = (const vfloat4*)in;
    vfloat4* __restrict__       out4 = (vfloat4*)out;

#pragma unroll
    for (int r = 0; r < ROWS_PER_BLOCK; ++r) {
        vfloat4 v = __builtin_nontemporal_load(&in4[idx]);
        v *= m;
        __builtin_nontemporal_store(v, &out4[idx]);
        idx += cstride4;
    }
}

// ---------------------------------------------------------------------------
extern "C" void kernel_launch(void* const* d_in, const int* in_sizes, int n_in,
                              void* d_out, int out_size, void* d_ws, size_t ws_size,
                              hipStream_t stream) {
    const float* in_data = (const float*)d_in[0];   // (8192, 4096) f32
    const float* fx      = (const float*)d_in[1];   // (64, 10000)  f32
    const float* fy      = (const float*)d_in[2];   // (64, 10000)  f32
    const int*   epoch   = (const int*)  d_in[3];   // scalar int
    float*       out     = (float*)d_out;

    const int total_pts = in_sizes[1] / N_FIBERS;          // 10000
    const int rows      = in_sizes[0] / INPUT_SIZE;        // 8192

    float* mask = (float*)d_ws;                            // 4096 floats scratch

    // 1) mask = 1.0
    mask_init_kernel<<<(INPUT_SIZE + TB - 1) / TB, TB, 0, stream>>>(mask);

    // 2) scatter 0.0 at touched cells (6400 points)
    const int P = N_FIBERS * PTS_PER_EPOCH;
    mask_scatter_kernel<<<(P + TB - 1) / TB, TB, 0, stream>>>(
        fx, fy, epoch, total_pts, mask);

    // 3) streaming multiply: grid = (col tiles, row groups)
    dim3 grid(INPUT_SIZE / COLS_PER_TILE, rows / ROWS_PER_BLOCK);
    mask_mul_kernel<<<grid, TB, 0, stream>>>(in_data, mask, out);
}